// GraphMAE_84507776516699
// MI455X (gfx1250) — compile-verified
//
#include <hip/hip_runtime.h>

typedef float v2f __attribute__((ext_vector_type(2)));
typedef float v8f __attribute__((ext_vector_type(8)));

#define IN_C 128
#define H_C  64

// A-transform modes for the WMMA GEMM
#define AMODE_PLAIN 0   // a = A[row][k]
#define AMODE_MASK  1   // a = mflag[row] ? mask_token[k] : A[row][k]   (pointer redirect)
#define AMODE_BRELU 2   // a = relu(A[row][k] + in_bias[k])             (bias staged in LDS)

// ---------------------------------------------------------------------------
__global__ void init_kernel(float* __restrict__ dinv, int* __restrict__ mflag, int n) {
    int i = blockIdx.x * blockDim.x + threadIdx.x;
    if (i < n) { dinv[i] = 1.0f; mflag[i] = 0; }
}

__global__ void mask_scatter_kernel(const int* __restrict__ mask_nodes,
                                    int* __restrict__ mflag, int mcnt) {
    int i = blockIdx.x * blockDim.x + threadIdx.x;
    if (i < mcnt) mflag[mask_nodes[i]] = 1;
}

__global__ void deg_kernel(const int* __restrict__ dst, float* __restrict__ dinv, int e) {
    int i = blockIdx.x * blockDim.x + threadIdx.x;
    if (i < e) unsafeAtomicAdd(&dinv[dst[i]], 1.0f);
}

__global__ void rsqrt_kernel(float* __restrict__ dinv, int n) {
    int i = blockIdx.x * blockDim.x + threadIdx.x;
    if (i < n) { float d = dinv[i]; dinv[i] = (d > 0.0f) ? rsqrtf(d) : 0.0f; }
}

__global__ void zero_kernel(float4* __restrict__ p, int n4) {
    int i = blockIdx.x * blockDim.x + threadIdx.x;
    if (i < n4) p[i] = make_float4(0.f, 0.f, 0.f, 0.f);
}

// ---------------------------------------------------------------------------
// WMMA GEMM:  D[M x NT*16] = transformA(A[M x K]) @ W[K x NT*16]  (+outbias,relu)
// wave32 per 16-row stripe; 4 waves / block.
// W is staged in LDS pre-swizzled so each lane's B fragment is a contiguous,
// 8B-aligned pair:  sW[((k>>2)*Nout + n)*4 + (k&3)] = W[k][n]
//   -> B frag (W[kb][n], W[kb+1][n]) = one ds_load_b64, no register repacking.
// f32 16x16x4 A-frag: lanes 0-15 -> (M=lane,   K=k0+{0,1})
//                     lanes16-31 -> (M=lane-16,K=k0+{2,3})
// C/D frag: VGPR r -> (M = r + (lane>=16 ? 8:0), N = lane&15)
template<int NT, int AMODE, bool OBIAS, bool ORELU>
__global__ __launch_bounds__(128) void gemm_wmma_kernel(
    const float* __restrict__ A, int lda,
    const float* __restrict__ Wg, int K,
    float* __restrict__ D, int ldo,
    const float* __restrict__ in_bias,
    const float* __restrict__ out_bias,
    const int* __restrict__ mflag, const float* __restrict__ mtok,
    int mtiles)
{
    extern __shared__ float smem[];
    const int Nout = NT * 16;
    float* sW = smem;                  // K x Nout, swizzled in 4-k groups
    float* sB = smem + K * Nout;       // K (only AMODE_BRELU)

    const int tid = threadIdx.x;
    for (int i = tid; i < K * Nout; i += 128) {
        const int k = i / Nout;        // Nout is a power of two -> shift
        const int n = i - k * Nout;
        sW[((k >> 2) * Nout + n) * 4 + (k & 3)] = Wg[i];
    }
    if (AMODE == AMODE_BRELU)
        for (int i = tid; i < K; i += 128) sB[i] = in_bias[i];
    __syncthreads();

    const int wave  = tid >> 5;
    const int lane  = tid & 31;
    const int mtile = blockIdx.x * 4 + wave;
    if (mtile >= mtiles) return;       // wave-uniform: EXEC all-ones around WMMA

    const int row   = mtile * 16 + (lane & 15);
    const int khalf = (lane >= 16) ? 2 : 0;
    const int ncol  = lane & 15;

    const float* arow = A + (size_t)row * lda;
    if (AMODE == AMODE_MASK && mflag[row] != 0) arow = mtok;  // branchless redirect

    // per-lane LDS base for B fragments (advances by Nout*4 floats per k-step)
    const float* bbase = sW + (size_t)ncol * 4 + khalf;

    v8f acc[NT];
#pragma unroll
    for (int nt = 0; nt < NT; ++nt)
        acc[nt] = (v8f){0.f,0.f,0.f,0.f,0.f,0.f,0.f,0.f};

#pragma unroll 4
    for (int k0 = 0; k0 < K; k0 += 4) {
        const int kb = k0 + khalf;
        v2f a = *(const v2f*)(arow + kb);                 // global_load_b64
        if (AMODE == AMODE_BRELU) {
            v2f bia = *(const v2f*)(sB + kb);             // ds_load_b64
            a.x = fmaxf(a.x + bia.x, 0.f);
            a.y = fmaxf(a.y + bia.y, 0.f);
        }
        const float* bk = bbase + (size_t)(k0 >> 2) * Nout * 4;
#pragma unroll
        for (int nt = 0; nt < NT; ++nt) {
            v2f b = *(const v2f*)(bk + nt * 64);          // ds_load_b64, imm offset
            acc[nt] = __builtin_amdgcn_wmma_f32_16x16x4_f32(
                false, a, false, b, (short)0, acc[nt], false, false);
        }
    }

#pragma unroll
    for (int nt = 0; nt < NT; ++nt) {
        const int n = nt * 16 + ncol;
        float ob = OBIAS ? out_bias[n] : 0.f;
#pragma unroll
        for (int r = 0; r < 8; ++r) {
            const int m = mtile * 16 + r + ((lane >= 16) ? 8 : 0);
            float v = acc[nt][r] + ob;
            if (ORELU) v = fmaxf(v, 0.f);
            D[(size_t)m * ldo + n] = v;
        }
    }
}

// ---------------------------------------------------------------------------
// Edge + self-loop scatter:  agg[dst] += norm * hw[src]
// 8 threads per entry, 8 contiguous floats each (two float4 gathers).
__global__ void scatter_kernel(const float* __restrict__ hw, float* __restrict__ agg,
                               const int* __restrict__ src, const int* __restrict__ dst,
                               const float* __restrict__ dinv, int e, int n)
{
    long t = (long)blockIdx.x * blockDim.x + threadIdx.x;
    long entry = t >> 3;
    if (entry >= (long)e + n) return;
    int c0 = (int)(t & 7) * 8;

    int s, d; float w;
    if (entry < e) {
        s = src[entry]; d = dst[entry];
        w = dinv[s] * dinv[d];
    } else {
        s = d = (int)(entry - e);
        float dv = dinv[s]; w = dv * dv;
    }
    const float4* hp = (const float4*)(hw + (size_t)s * H_C + c0);
    float4 v0 = hp[0];
    float4 v1 = hp[1];
    float* ap = agg + (size_t)d * H_C + c0;
    unsafeAtomicAdd(ap + 0, v0.x * w);
    unsafeAtomicAdd(ap + 1, v0.y * w);
    unsafeAtomicAdd(ap + 2, v0.z * w);
    unsafeAtomicAdd(ap + 3, v0.w * w);
    unsafeAtomicAdd(ap + 4, v1.x * w);
    unsafeAtomicAdd(ap + 5, v1.y * w);
    unsafeAtomicAdd(ap + 6, v1.z * w);
    unsafeAtomicAdd(ap + 7, v1.w * w);
}

// ---------------------------------------------------------------------------
// out_x[m] = x[mask_nodes[m]]  (output part 1), and compact decoder input
// zc[m] = relu(agg2[mask_nodes[m]] + b2).  float4 lanes: 32 per row.
__global__ void gather_kernel(const float4* __restrict__ x4, const float4* __restrict__ agg4,
                              const float4* __restrict__ b2_4, const int* __restrict__ mask_nodes,
                              float4* __restrict__ out_x4, float4* __restrict__ zc4, int mcnt)
{
    long t = (long)blockIdx.x * blockDim.x + threadIdx.x;
    if (t >= (long)mcnt * 32) return;
    int m  = (int)(t >> 5);
    int c4 = (int)(t & 31);
    int node = mask_nodes[m];
    out_x4[(size_t)m * 32 + c4] = x4[(size_t)node * 32 + c4];
    if (c4 < 16) {
        float4 a = agg4[(size_t)node * 16 + c4];
        float4 b = b2_4[c4];
        float4 z;
        z.x = fmaxf(a.x + b.x, 0.f);
        z.y = fmaxf(a.y + b.y, 0.f);
        z.z = fmaxf(a.z + b.z, 0.f);
        z.w = fmaxf(a.w + b.w, 0.f);
        zc4[(size_t)m * 16 + c4] = z;
    }
}

// ---------------------------------------------------------------------------
extern "C" void kernel_launch(void* const* d_in, const int* in_sizes, int n_in,
                              void* d_out, int out_size, void* d_ws, size_t ws_size,
                              hipStream_t stream)
{
    const float* x      = (const float*)d_in[0];
    const int*   edge   = (const int*)d_in[1];
    const int*   mnodes = (const int*)d_in[2];
    const float* W0 = (const float*)d_in[3];  const float* b0 = (const float*)d_in[4];
    const float* W1 = (const float*)d_in[5];  const float* b1 = (const float*)d_in[6];
    const float* W2 = (const float*)d_in[7];  const float* b2 = (const float*)d_in[8];
    const float* D0w = (const float*)d_in[9];  const float* D0b = (const float*)d_in[10];
    const float* D1w = (const float*)d_in[11]; const float* D1b = (const float*)d_in[12];
    const float* D2w = (const float*)d_in[13]; const float* D2b = (const float*)d_in[14];
    const float* mtok = (const float*)d_in[15];

    const int N  = in_sizes[0] / IN_C;   // 100000
    const int E  = in_sizes[1] / 2;      // 1600000
    const int MK = in_sizes[2];          // 50000
    const int* src = edge;
    const int* dst = edge + E;

    // --- workspace carve-out (512B aligned) ---
    char* w = (char*)d_ws;
    auto carve = [&](size_t bytes) -> char* {
        char* p = w; w += (bytes + 511) & ~(size_t)511; return p;
    };
    float* dinv  = (float*)carve((size_t)N * 4);
    int*   mflag = (int*)  carve((size_t)N * 4);
    float* hw    = (float*)carve((size_t)N * H_C * 4);
    float* agg   = (float*)carve((size_t)N * H_C * 4);
    float* zc    = (float*)carve((size_t)MK * H_C * 4);
    float* rc0   = (float*)carve((size_t)MK * H_C * 4);
    float* rc1   = (float*)carve((size_t)MK * H_C * 4);

    const int B = 256;
    // --- GCN normalization prep ---
    init_kernel<<<(N + B - 1) / B, B, 0, stream>>>(dinv, mflag, N);
    mask_scatter_kernel<<<(MK + B - 1) / B, B, 0, stream>>>(mnodes, mflag, MK);
    deg_kernel<<<(E + B - 1) / B, B, 0, stream>>>(dst, dinv, E);
    rsqrt_kernel<<<(N + B - 1) / B, B, 0, stream>>>(dinv, N);

    const int mt  = N / 16;              // 6250 row tiles (exact)
    const int gm  = (mt + 3) / 4;
    const int n4  = N * H_C / 4;
    const unsigned sc = (unsigned)(((long)(E + N) * 8 + B - 1) / B);

    // --- encoder layer 0: hw = mask(x) @ W0 ---
    gemm_wmma_kernel<4, AMODE_MASK, false, false>
        <<<gm, 128, IN_C * H_C * 4, stream>>>(
        x, IN_C, W0, IN_C, hw, H_C, nullptr, nullptr, mflag, mtok, mt);
    zero_kernel<<<(n4 + B - 1) / B, B, 0, stream>>>((float4*)agg, n4);
    scatter_kernel<<<sc, B, 0, stream>>>(hw, agg, src, dst, dinv, E, N);

    // --- encoder layer 1: hw = relu(agg + b0) @ W1 ---
    gemm_wmma_kernel<4, AMODE_BRELU, false, false>
        <<<gm, 128, (H_C * H_C + H_C) * 4, stream>>>(
        agg, H_C, W1, H_C, hw, H_C, b0, nullptr, nullptr, nullptr, mt);
    zero_kernel<<<(n4 + B - 1) / B, B, 0, stream>>>((float4*)agg, n4);
    scatter_kernel<<<sc, B, 0, stream>>>(hw, agg, src, dst, dinv, E, N);

    // --- encoder layer 2: hw = relu(agg + b1) @ W2 ---
    gemm_wmma_kernel<4, AMODE_BRELU, false, false>
        <<<gm, 128, (H_C * H_C + H_C) * 4, stream>>>(
        agg, H_C, W2, H_C, hw, H_C, b1, nullptr, nullptr, nullptr, mt);
    zero_kernel<<<(n4 + B - 1) / B, B, 0, stream>>>((float4*)agg, n4);
    scatter_kernel<<<sc, B, 0, stream>>>(hw, agg, src, dst, dinv, E, N);

    // --- output part 1 + compact z = relu(agg + b2) at masked rows ---
    float* out_x = (float*)d_out;
    float* out_r = (float*)d_out + (size_t)MK * IN_C;
    gather_kernel<<<(int)(((long)MK * 32 + B - 1) / B), B, 0, stream>>>(
        (const float4*)x, (const float4*)agg, (const float4*)b2,
        mnodes, (float4*)out_x, (float4*)zc, MK);

    // --- decoder (masked rows only): 3 GEMMs, last writes d_out directly ---
    const int mt2 = MK / 16;             // 3125 tiles (exact)
    const int gm2 = (mt2 + 3) / 4;
    gemm_wmma_kernel<4, AMODE_PLAIN, true, true>
        <<<gm2, 128, H_C * H_C * 4, stream>>>(
        zc, H_C, D0w, H_C, rc0, H_C, nullptr, D0b, nullptr, nullptr, mt2);
    gemm_wmma_kernel<4, AMODE_PLAIN, true, true>
        <<<gm2, 128, H_C * H_C * 4, stream>>>(
        rc0, H_C, D1w, H_C, rc1, H_C, nullptr, D1b, nullptr, nullptr, mt2);
    gemm_wmma_kernel<8, AMODE_PLAIN, true, false>
        <<<gm2, 128, H_C * IN_C * 4, stream>>>(
        rc1, H_C, D2w, H_C, out_r, IN_C, nullptr, D2b, nullptr, nullptr, mt2);
}